// Encoder_83846351553267
// MI455X (gfx1250) — compile-verified
//
#include <hip/hip_runtime.h>
#include <hip/hip_bf16.h>

typedef float v2f __attribute__((ext_vector_type(2)));
typedef float v8f __attribute__((ext_vector_type(8)));

#define BB 4  // batch

// ---------------------------------------------------------------- conv2d 7x7 s2 p3
__global__ void conv2d7(const float* __restrict__ x, const float* __restrict__ w,
                        float* __restrict__ y, int IC, int H, int W, int OC) {
  int OH = H >> 1, OW = W >> 1;
  int i = blockIdx.x * blockDim.x + threadIdx.x;
  int total = BB * OC * OH * OW;
  if (i >= total) return;
  int ow = i % OW; int t = i / OW;
  int oh = t % OH; t /= OH;
  int oc = t % OC; int b = t / OC;
  float acc = 0.f;
  for (int ic = 0; ic < IC; ++ic)
    for (int kh = 0; kh < 7; ++kh) {
      int ih = 2 * oh + kh - 3;
      if (ih < 0 || ih >= H) continue;
      for (int kw = 0; kw < 7; ++kw) {
        int iw = 2 * ow + kw - 3;
        if (iw < 0 || iw >= W) continue;
        acc += x[(((size_t)b * IC + ic) * H + ih) * W + iw] *
               w[((oc * IC + ic) * 7 + kh) * 7 + kw];
      }
    }
  y[i] = acc;
}

// ---------------------------------------------------------------- conv1d k7 s2 p3
__global__ void conv1d7(const float* __restrict__ x, const float* __restrict__ w,
                        float* __restrict__ y, int IC, int Lin, int OC) {
  int Lout = Lin >> 1;
  int i = blockIdx.x * blockDim.x + threadIdx.x;
  int total = BB * OC * Lout;
  if (i >= total) return;
  int ol = i % Lout; int t = i / Lout;
  int oc = t % OC; int b = t / OC;
  float acc = 0.f;
  for (int ic = 0; ic < IC; ++ic) {
    const float* xr = x + ((size_t)b * IC + ic) * Lin;
    const float* wr = w + (size_t)(oc * IC + ic) * 7;
    for (int k = 0; k < 7; ++k) {
      int il = 2 * ol + k - 3;
      if (il >= 0 && il < Lin) acc += xr[il] * wr[k];
    }
  }
  y[i] = acc;
}

// ---------------------------------------------------------------- transposes
// t (B, C, L) -> u (B*L, C)
__global__ void t2u(const float* __restrict__ t, float* __restrict__ u, int C, int L) {
  int i = blockIdx.x * blockDim.x + threadIdx.x;
  int total = BB * C * L;
  if (i >= total) return;
  int c = i % C; int m = i / C; int l = m % L; int b = m / L;
  u[i] = t[((size_t)b * C + c) * L + l];
}
// u (B*L, C) -> t (B, C, L)
__global__ void u2t(const float* __restrict__ u, float* __restrict__ t, int C, int L) {
  int i = blockIdx.x * blockDim.x + threadIdx.x;
  int total = BB * C * L;
  if (i >= total) return;
  int l = i % L; int r = i / L; int c = r % C; int b = r / C;
  t[i] = u[((size_t)(b * L + l)) * C + c];
}

// ---------------------------------------------------------------- WMMA f32 GEMM
// C[M,N] = A[M,K] * W[N,K]^T   (nn.Linear convention), M%16==0, K%4==0.
// One wave computes one 16x16 tile via V_WMMA_F32_16X16X4_F32.
// N edge: column n of D depends only on column n of B, so out-of-range lanes
// just CLAMP their W row (their products land in columns that are never stored).
// This keeps EXEC all-ones and makes both fragment loads plain global_load_b64.
__global__ void gemm_wmma(const float* __restrict__ A, const float* __restrict__ W,
                          float* __restrict__ C, int M, int N, int K) {
  int wave = threadIdx.x >> 5;
  int lane = threadIdx.x & 31;
  int n0 = (blockIdx.x * 4 + wave) * 16;
  int m0 = blockIdx.y * 16;
  if (n0 >= N) return;              // wave-uniform: EXEC stays all-ones for WMMA
  int r = lane & 15, g = lane >> 4; // frag row/col = r, K = k0 + 2*g + vgpr
  const float* Arow = A + (size_t)(m0 + r) * K + 2 * g;
  int ncol = n0 + r;
  int nclamp = (ncol < N) ? ncol : (N - 1);
  const float* Wrow = W + (size_t)nclamp * K + 2 * g;
  v8f acc = {0.f, 0.f, 0.f, 0.f, 0.f, 0.f, 0.f, 0.f};
#if defined(__gfx1250__) && __has_builtin(__builtin_amdgcn_wmma_f32_16x16x4_f32)
#pragma unroll 8
  for (int k = 0; k < K; k += 4) {
    v2f a = *(const v2f*)(Arow + k);   // 8B aligned: K%4==0, k%4==0
    v2f b = *(const v2f*)(Wrow + k);
    acc = __builtin_amdgcn_wmma_f32_16x16x4_f32(false, a, false, b,
                                                (short)0, acc, false, false);
  }
#else
  // per-lane fallback with identical output mapping
  for (int v = 0; v < 8; ++v) {
    const float* ar = A + (size_t)(m0 + v + 8 * g) * K;
    const float* wr = W + (size_t)nclamp * K;
    float s = 0.f;
    for (int k = 0; k < K; ++k) s += ar[k] * wr[k];
    acc[v] = s;
  }
#endif
#pragma unroll
  for (int v = 0; v < 8; ++v) {
    int row = m0 + v + 8 * g;
    if (ncol < N) C[(size_t)row * N + ncol] = acc[v];
  }
}

// ---------------------------------------------------------------- depthwise conv4 + SiLU
__global__ void dwconv_silu(const float* __restrict__ Z, const float* __restrict__ w,
                            const float* __restrict__ bias, float* __restrict__ out,
                            int L, int nout, int d_inner, int conv_dim) {
  int i = blockIdx.x * blockDim.x + threadIdx.x;
  int total = BB * L * conv_dim;
  if (i >= total) return;
  int c = i % conv_dim; int m = i / conv_dim;
  int l = m % L; int b = m / L;
  float acc = bias[c];
  const float* wc = w + c * 4;
#pragma unroll
  for (int j = 0; j < 4; ++j) {
    int l2 = l + j - 3;
    if (l2 >= 0) acc += wc[j] * Z[((size_t)(b * L + l2)) * nout + d_inner + c];
  }
  out[i] = acc / (1.f + __expf(-acc));  // SiLU
}

// ---------------------------------------------------------------- dt softplus + decay
__global__ void dtprep(const float* __restrict__ Z, const float* __restrict__ dt_bias,
                       const float* __restrict__ A_log, float* __restrict__ dt,
                       float* __restrict__ dec, int M, int nout, int nheads) {
  int i = blockIdx.x * blockDim.x + threadIdx.x;
  if (i >= M * nheads) return;
  int h = i % nheads; int m = i / nheads;
  float raw = Z[(size_t)m * nout + (nout - nheads) + h] + dt_bias[h];
  float d = (raw > 20.f) ? raw : log1pf(__expf(raw));
  dt[i] = d;
  dec[i] = __expf(-d * __expf(A_log[h]));
}

// ---------------------------------------------------------------- SSM sequential scan
// grid (nheads, B), block 256.  State (64 x 128) f32 held in VGPRs: 32/lane.
__global__ void ssm_scan(const float* __restrict__ xbc, const float* __restrict__ dtv,
                         const float* __restrict__ dec, const float* __restrict__ Dp,
                         float* __restrict__ y, int L, int nheads, int d_inner,
                         int conv_dim) {
  int h = blockIdx.x, b = blockIdx.y;
  int t = threadIdx.x;
  int p = t & 63, chunk = t >> 6;
  __shared__ float sB[128], sC[128], sx[64], red[256];
  float st[32];
#pragma unroll
  for (int i = 0; i < 32; ++i) st[i] = 0.f;
  float Dh = Dp[h];
  for (int l = 0; l < L; ++l) {
    int m = b * L + l;
    const float* row = xbc + (size_t)m * conv_dim;
    if (t < 128) { sB[t] = row[d_inner + t]; sC[t] = row[d_inner + 128 + t]; }
    if (t < 64)  { sx[t] = row[h * 64 + t]; }
    __syncthreads();
    float dv    = dec[(size_t)m * nheads + h];
    float coeff = dtv[(size_t)m * nheads + h] * sx[p];
    const float* Bn = sB + chunk * 32;
    const float* Cn = sC + chunk * 32;
    float part = 0.f;
#pragma unroll
    for (int i = 0; i < 32; ++i) {
      st[i] = st[i] * dv + coeff * Bn[i];
      part += st[i] * Cn[i];
    }
    red[t] = part;
    __syncthreads();
    if (t < 64) {
      float yv = red[t] + red[64 + t] + red[128 + t] + red[192 + t] + Dh * sx[t];
      y[(size_t)m * d_inner + h * 64 + t] = yv;
    }
    __syncthreads();
  }
}

// ---------------------------------------------------------------- gate * SiLU(z) + RMSNorm
__global__ void gate_rmsnorm(const float* __restrict__ y, const float* __restrict__ Z,
                             const float* __restrict__ nw, float* __restrict__ out,
                             int d_inner, int nout) {
  int m = blockIdx.x; int t = threadIdx.x;
  __shared__ float ssum[256];
  float acc = 0.f;
  for (int c = t; c < d_inner; c += 256) {
    float zv = Z[(size_t)m * nout + c];
    float g = y[(size_t)m * d_inner + c] * (zv / (1.f + __expf(-zv)));
    out[(size_t)m * d_inner + c] = g;
    acc += g * g;
  }
  ssum[t] = acc;
  __syncthreads();
  for (int s = 128; s > 0; s >>= 1) {
    if (t < s) ssum[t] += ssum[t + s];
    __syncthreads();
  }
  float r = rsqrtf(ssum[0] / (float)d_inner + 1e-5f);
  for (int c = t; c < d_inner; c += 256)
    out[(size_t)m * d_inner + c] *= r * nw[c];
}

// ================================================================ host side
struct BlkP { const float *A_log, *D, *conv_b, *conv_w, *dt_bias, *in_proj, *norm_w, *out_proj; };
struct LayerP { const float* conv_w; BlkP blk[2]; };

extern "C" void kernel_launch(void* const* d_in, const int* in_sizes, int n_in,
                              void* d_out, int out_size, void* d_ws, size_t ws_size,
                              hipStream_t stream) {
  (void)out_size; (void)ws_size;
  // -------- locate tokens by unique size
  int tok = -1;
  for (int i = 0; i < n_in; ++i)
    if (in_sizes[i] == 4 * 1 * 128 * 2048) { tok = i; break; }
  if (tok < 0 || n_in < 10) return;
  const float* tokens = (const float*)d_in[tok];

  const float* leaf[64]; int lsz[64]; int nleaf = 0;
  for (int i = 0; i < n_in && nleaf < 64; ++i)
    if (i != tok) { leaf[nleaf] = (const float*)d_in[i]; lsz[nleaf] = in_sizes[i]; ++nleaf; }
  if (nleaf < 53) return;

  const float* conv1w = leaf[0];   // (2,1,7,7)
  const float* conv2w = leaf[1];   // (4,2,7,7)
  LayerP lp[3];
  int pos = 2;
  for (int l = 0; l < 3; ++l) {
    bool sortedOrder = (lsz[pos] <= 64);  // jax tree order: block A_log (8/16/32) first
    if (sortedOrder) {
      for (int j = 0; j < 2; ++j) {
        BlkP& bk = lp[l].blk[j];
        bk.A_log = leaf[pos++]; bk.D = leaf[pos++]; bk.conv_b = leaf[pos++];
        bk.conv_w = leaf[pos++]; bk.dt_bias = leaf[pos++]; bk.in_proj = leaf[pos++];
        bk.norm_w = leaf[pos++]; bk.out_proj = leaf[pos++];
      }
      lp[l].conv_w = leaf[pos++];
    } else {                               // insertion order: layer conv_w first
      lp[l].conv_w = leaf[pos++];
      for (int j = 0; j < 2; ++j) {
        BlkP& bk = lp[l].blk[j];
        bk.in_proj = leaf[pos++]; bk.conv_w = leaf[pos++]; bk.conv_b = leaf[pos++];
        bk.dt_bias = leaf[pos++]; bk.A_log = leaf[pos++]; bk.D = leaf[pos++];
        bk.norm_w = leaf[pos++]; bk.out_proj = leaf[pos++];
      }
    }
  }

  // -------- workspace carve (floats); total ~17.9 MB
  float* ws = (float*)d_ws;
  size_t o = 0;
  float* bufT1  = ws + o; o += 524288;   // conv1 out / layer conv out
  float* bufT2  = ws + o; o += 262144;   // conv2 out / t between layers
  float* bufU0  = ws + o; o += 262144;
  float* bufU1  = ws + o; o += 262144;
  float* bufZ   = ws + o; o += 1318912;  // zxbcdt (M x nout), sized for layer 0 (largest)
  float* bufXBC = ws + o; o += 786432;
  float* bufDT  = ws + o; o += 8192;
  float* bufDec = ws + o; o += 8192;
  float* bufY   = ws + o; o += 524288;
  float* bufYN  = ws + o; o += 524288;

  // -------- stem
  {
    int total = BB * 2 * 64 * 1024;
    conv2d7<<<(total + 255) / 256, 256, 0, stream>>>(tokens, conv1w, bufT1, 1, 128, 2048, 2);
  }
  {
    int total = BB * 4 * 32 * 512;
    conv2d7<<<(total + 255) / 256, 256, 0, stream>>>(bufT1, conv2w, bufT2, 2, 64, 1024, 4);
  }
  // bufT2 == t (4, 128, 512) after implicit reshape (contiguous view)

  for (int l = 0; l < 3; ++l) {
    int Cin = 128 << l, dm = 256 << l;
    int Lin = 512 >> l, L = 256 >> l;
    int d_inner = 2 * dm, nheads = dm / 32;
    int nout = 2 * d_inner + 256 + nheads;
    int conv_dim = d_inner + 256;
    int M = BB * L;

    {
      int total = BB * dm * L;
      conv1d7<<<(total + 255) / 256, 256, 0, stream>>>(bufT2, lp[l].conv_w, bufT1, Cin, Lin, dm);
      t2u<<<(total + 255) / 256, 256, 0, stream>>>(bufT1, bufU0, dm, L);
    }

    float* uin = bufU0;
    float* uout = bufU1;
    for (int j = 0; j < 2; ++j) {
      const BlkP& bk = lp[l].blk[j];
      // in_proj GEMM: (M x dm) x (nout x dm)^T -> (M x nout)
      {
        dim3 grid((nout + 63) / 64, M / 16);
        gemm_wmma<<<grid, 128, 0, stream>>>(uin, bk.in_proj, bufZ, M, nout, dm);
      }
      // depthwise causal conv4 + bias + SiLU
      {
        int total = M * conv_dim;
        dwconv_silu<<<(total + 255) / 256, 256, 0, stream>>>(bufZ, bk.conv_w, bk.conv_b,
                                                             bufXBC, L, nout, d_inner, conv_dim);
      }
      // dt softplus + decay
      {
        int total = M * nheads;
        dtprep<<<(total + 255) / 256, 256, 0, stream>>>(bufZ, bk.dt_bias, bk.A_log,
                                                        bufDT, bufDec, M, nout, nheads);
      }
      // sequential SSM scan, state in VGPRs
      {
        dim3 grid(nheads, BB);
        ssm_scan<<<grid, 256, 0, stream>>>(bufXBC, bufDT, bufDec, bk.D, bufY,
                                           L, nheads, d_inner, conv_dim);
      }
      // y * silu(z), RMSNorm
      gate_rmsnorm<<<M, 256, 0, stream>>>(bufY, bufZ, bk.norm_w, bufYN, d_inner, nout);
      // out_proj GEMM: (M x d_inner) x (dm x d_inner)^T -> (M x dm)
      {
        dim3 grid((dm + 63) / 64, M / 16);
        gemm_wmma<<<grid, 128, 0, stream>>>(bufYN, bk.out_proj, uout, M, dm, d_inner);
      }
      float* tmp = uin; uin = uout; uout = tmp;
    }
    // back to (B, dm, L); final layer writes d_out directly
    float* tdst = (l == 2) ? (float*)d_out : bufT2;
    int total = BB * dm * L;
    u2t<<<(total + 255) / 256, 256, 0, stream>>>(uin, tdst, dm, L);
  }
}